// STDWetDry_12893491823246
// MI455X (gfx1250) — compile-verified
//
#include <hip/hip_runtime.h>

// Problem constants (match reference setup_inputs / N_STEPS / TH)
#define B_ROWS   256
#define L_LEN    65536
#define NWIN     32
#define PAD_BEG  15          // (NWIN-1)/2
#define VAR_TH   1.21f       // sigma > 1.1  <=>  var > 1.21

// Tiling
#define THREADS  256
#define COUT     32                    // outputs per thread (rolling window)
#define TILE     (THREADS * COUT)      // 8192 outputs per block
#define LDS_LOG  (TILE + 32)           // logical floats staged (halo 16 each side)

// Padded LDS index: +4 dwords every 64. Keeps 4-dword groups contiguous and
// 16B aligned (B128-safe) while making the per-lane compute stride
// 32t + 4*(t/2), which still puts some lanes on one bank (e.g. t=1 and t=16).
__device__ __forceinline__ int pidx(int i) { return i + ((i >> 6) << 2); }
#define LDS_PHYS (LDS_LOG + ((LDS_LOG >> 6) << 2) + 4)

typedef int   v4i __attribute__((ext_vector_type(4)));
typedef float v4f __attribute__((ext_vector_type(4)));
typedef __attribute__((address_space(1))) int gint;   // global int
typedef __attribute__((address_space(3))) int lint;   // LDS int
typedef __attribute__((address_space(1))) v4i gv4i;   // global 4xint
typedef __attribute__((address_space(3))) v4i lv4i;   // LDS 4xint

#if __has_builtin(__builtin_amdgcn_global_load_async_to_lds_b32)
#define HAVE_ASYNC_LDS 1
#else
#define HAVE_ASYNC_LDS 0
#endif
#if __has_builtin(__builtin_amdgcn_global_load_async_to_lds_b128)
#define HAVE_ASYNC_LDS128 1
#else
#define HAVE_ASYNC_LDS128 0
#endif

__device__ __forceinline__ void stage_one(const float* gp, float* lp) {
#if HAVE_ASYNC_LDS
    __builtin_amdgcn_global_load_async_to_lds_b32(
        (gint*)(float*)gp, (lint*)lp, /*offset=*/0, /*cpol=*/0);
#else
    *lp = *gp;
#endif
}

__device__ __forceinline__ void stage_four(const float* gp, float* lp) {
#if HAVE_ASYNC_LDS128
    __builtin_amdgcn_global_load_async_to_lds_b128(
        (gv4i*)(float*)gp, (lv4i*)lp, /*offset=*/0, /*cpol=*/0);
#else
    stage_one(gp + 0, lp + 0);
    stage_one(gp + 1, lp + 1);
    stage_one(gp + 2, lp + 2);
    stage_one(gp + 3, lp + 3);
#endif
}

__device__ __forceinline__ void async_wait_all() {
#if HAVE_ASYNC_LDS
#if __has_builtin(__builtin_amdgcn_s_wait_asynccnt)
    __builtin_amdgcn_s_wait_asynccnt(0);
#else
    asm volatile("s_wait_asynccnt 0" ::: "memory");
#endif
#endif
}

__device__ __forceinline__ float fast_sqrtf(float v) {
#if __has_builtin(__builtin_amdgcn_sqrtf)
    return __builtin_amdgcn_sqrtf(v);   // single v_sqrt_f32
#else
    return sqrtf(v);
#endif
}

__global__ __launch_bounds__(THREADS) void std_wetdry_kernel(
    const float* __restrict__ x,     // [B, L]
    float* __restrict__ out_res,     // [B, L]  clip(round(sigma/2.2),0,1)
    float* __restrict__ out_sig)     // [B, L]  sigma_n_base
{
    __shared__ float lds[LDS_PHYS];

    const int tid      = threadIdx.x;
    const int row      = blockIdx.y;
    const int tileBase = blockIdx.x * TILE;
    const float* xrow  = x + (size_t)row * L_LEN;
    const bool  edge   = (blockIdx.x == 0) | (blockIdx.x == gridDim.x - 1);

    // ---- Stage x[tileBase-16 .. tileBase+TILE+15] into padded LDS (async) ----
    if (!edge) {
        // Interior tile: b128 chunks, no clamping. LDS_LOG/4 = 2056 chunks.
        const float* gbase = xrow + (tileBase - 16);
#pragma unroll
        for (int k = 0; k < (LDS_LOG / 4) / THREADS; ++k) {      // 8 full passes
            const int c4 = tid + k * THREADS;
            stage_four(gbase + 4 * c4, &lds[pidx(4 * c4)]);
        }
        if (tid < (LDS_LOG / 4) % THREADS) {                     // 8 leftover chunks
            const int c4 = tid + ((LDS_LOG / 4) / THREADS) * THREADS;
            stage_four(gbase + 4 * c4, &lds[pidx(4 * c4)]);
        }
    } else {
        // Edge tile: b32 with clamp; clamped halo only feeds outputs forced to 0.
        for (int i = tid; i < LDS_LOG; i += THREADS) {
            int g = tileBase - 16 + i;
            g = (g < 0) ? 0 : ((g > L_LEN - 1) ? (L_LEN - 1) : g);
            stage_one(xrow + g, &lds[pidx(i)]);
        }
    }
    async_wait_all();
    __syncthreads();

    // ---- Rolling-window std: thread t owns outputs o = t*COUT .. +COUT-1 ----
    // Window for output o is LDS logical [o+1, o+32] (logical 0 == x[tileBase-16]).
    const int o0 = tid * COUT;

    float s1 = 0.0f, s2 = 0.0f;
#pragma unroll
    for (int q = 1; q <= NWIN; ++q) {
        const float v = lds[pidx(o0 + q)];
        s1 += v;
        s2 = fmaf(v, v, s2);
    }

    float sig_buf[COUT];
    float res_buf[COUT];
    const float inv_n = 1.0f / (float)NWIN;

#pragma unroll
    for (int c = 0; c < COUT; ++c) {
        if (c > 0) {
            const float vn = lds[pidx(o0 + c + NWIN)];
            const float vo = lds[pidx(o0 + c)];
            const float d  = vn - vo;
            s1 += d;
            s2 = fmaf(d, vn + vo, s2);       // vn^2 - vo^2
        }
        const float mean = s1 * inv_n;
        float var = fmaf(-mean, mean, s2 * inv_n);
        var = fmaxf(var, 0.0f);
        res_buf[c] = (var > VAR_TH) ? 1.0f : 0.0f;   // off the sqrt latency chain
        sig_buf[c] = fast_sqrtf(var);
    }

    // Reference zero-pads: sigma_n_base[j] == 0 for j < 15 or j >= L-16.
    if (edge) {
#pragma unroll
        for (int c = 0; c < COUT; ++c) {
            const int j = tileBase + o0 + c;
            if (j < PAD_BEG || j >= L_LEN - 16) { sig_buf[c] = 0.0f; res_buf[c] = 0.0f; }
        }
    }

    // ---- Streaming (non-temporal) 128-bit stores; 128B-aligned per thread ----
    const size_t obase = (size_t)row * L_LEN + (size_t)(tileBase + o0);
    v4f* r4 = (v4f*)(out_res + obase);
    v4f* s4 = (v4f*)(out_sig + obase);
#pragma unroll
    for (int k = 0; k < COUT / 4; ++k) {
        const v4f sv = { sig_buf[4*k+0], sig_buf[4*k+1],
                         sig_buf[4*k+2], sig_buf[4*k+3] };
        const v4f rv = { res_buf[4*k+0], res_buf[4*k+1],
                         res_buf[4*k+2], res_buf[4*k+3] };
        __builtin_nontemporal_store(rv, r4 + k);
        __builtin_nontemporal_store(sv, s4 + k);
    }
}

extern "C" void kernel_launch(void* const* d_in, const int* in_sizes, int n_in,
                              void* d_out, int out_size, void* d_ws, size_t ws_size,
                              hipStream_t stream) {
    (void)in_sizes; (void)n_in; (void)out_size; (void)d_ws; (void)ws_size;
    const float* x = (const float*)d_in[0];
    float* out     = (float*)d_out;
    float* out_res = out;                                  // first output tensor
    float* out_sig = out + (size_t)B_ROWS * (size_t)L_LEN; // second output tensor

    dim3 grid(L_LEN / TILE, B_ROWS, 1);
    dim3 block(THREADS, 1, 1);
    std_wetdry_kernel<<<grid, block, 0, stream>>>(x, out_res, out_sig);
}